// Virtual_Node_2422361555232
// MI455X (gfx1250) — compile-verified
//
#include <hip/hip_runtime.h>

// Problem constants (setup_inputs is fixed: N=1e6, D=128, G=1024).
#define DD 128
#define GMAX 1024

typedef float v2f __attribute__((ext_vector_type(2)));
typedef float v4f __attribute__((ext_vector_type(4)));
typedef float v8f __attribute__((ext_vector_type(8)));

// ---------------------------------------------------------------- kernel 0
// Zero the segment-sum accumulator (G*D floats). Deterministic re-init each call.
__global__ void k_zero(float* __restrict__ y, const int* __restrict__ gptr) {
    const int n = gptr[0] * DD;
    const int i = blockIdx.x * blockDim.x + threadIdx.x;
    if (i < n) y[i] = 0.0f;
}

// ---------------------------------------------------------------- kernel 1
// Sorted segment sum. One wave = one full 128-col row (32 lanes x float4).
// Register-accumulate runs of equal batch_idx; atomically flush at boundaries.
__global__ void k_segsum(const float* __restrict__ H, const int* __restrict__ idx,
                         float* __restrict__ y, int N) {
    const int lane   = threadIdx.x & 31;
    const int wavesB = blockDim.x >> 5;
    const int gwave  = blockIdx.x * wavesB + (threadIdx.x >> 5);
    const int nwaves = gridDim.x * wavesB;
    const int rowsPer = (N + nwaves - 1) / nwaves;
    int r0 = gwave * rowsPer;
    int r1 = r0 + rowsPer; if (r1 > N) r1 = N;
    if (r0 >= r1) return;

    const int c = lane * 4;
    v4f acc = {0.f, 0.f, 0.f, 0.f};
    int gprev = idx[r0];

    for (int r = r0; r < r1; ++r) {
        const int g = idx[r];                 // uniform within wave -> 1 request
        if (g != gprev) {                     // segment boundary: flush
            float* dst = y + (size_t)gprev * DD + c;
            atomicAdd(dst + 0, acc.x);
            atomicAdd(dst + 1, acc.y);
            atomicAdd(dst + 2, acc.z);
            atomicAdd(dst + 3, acc.w);
            acc = (v4f){0.f, 0.f, 0.f, 0.f};
            gprev = g;
        }
        // H is streamed once here, once in k3: non-temporal (TH=NT) keeps
        // y / idx / W hot in L2 instead.
        const v4f h =
            __builtin_nontemporal_load((const v4f*)(H + (size_t)r * DD) + lane);
        acc += h;
    }
    float* dst = y + (size_t)gprev * DD + c;
    atomicAdd(dst + 0, acc.x);
    atomicAdd(dst + 1, acc.y);
    atomicAdd(dst + 2, acc.z);
    atomicAdd(dst + 3, acc.w);
}

// ---------------------------------------------------------------- kernel 2
// y2 = relu(y @ W + b) via V_WMMA_F32_16X16X4_F32 (full f32 precision).
// One wave computes one 16x16 output tile; K=128 -> 32 WMMA issues.
// A 16x4 layout (ISA 7.12.2): lanes 0-15 hold K=0(v0),1(v1); lanes 16-31 K=2,3.
// B 4x16 mirrored; C/D: vgpr v, lanes 0-15 -> M=v, lanes 16-31 -> M=v+8, N=lane%16.
__global__ void k_gemm_relu(const float* __restrict__ Y, const float* __restrict__ W,
                            const float* __restrict__ bias, float* __restrict__ Y2,
                            const int* __restrict__ gptr) {
    const int G  = gptr[0];
    const int NT = DD / 16;                       // 8 tiles along N
    const int MT = G / 16;                        // tiles along M
    const int gw = blockIdx.x * (blockDim.x >> 5) + (threadIdx.x >> 5);
    if (gw >= MT * NT) return;                    // wave-uniform: EXEC stays full

    const int mt   = gw / NT;
    const int nt   = gw % NT;
    const int lane = threadIdx.x & 31;
    const int l15  = lane & 15;                   // row for A, col for B/C
    const int half = lane >> 4;                   // 0: K+{0,1}, 1: K+{2,3}

    const float* arow = Y + (size_t)(mt * 16 + l15) * DD;
    const float* bcol = W + nt * 16 + l15;

    v8f c = {};
    #pragma unroll 4
    for (int kb = 0; kb < DD; kb += 4) {
        const int k0 = kb + 2 * half;
        v2f a, b;
        a.x = arow[k0];
        a.y = arow[k0 + 1];
        b.x = bcol[(size_t)k0 * DD];
        b.y = bcol[(size_t)(k0 + 1) * DD];
        c = __builtin_amdgcn_wmma_f32_16x16x4_f32(
                /*neg_a=*/false, a, /*neg_b=*/false, b,
                /*c_mod=*/(short)0, c, /*reuse_a=*/false, /*reuse_b=*/false);
    }

    const int n  = nt * 16 + l15;
    const float bb = bias[n];
    #pragma unroll
    for (int v = 0; v < 8; ++v) {
        const int m = mt * 16 + v + half * 8;
        float val = c[v] + bb;
        Y2[(size_t)m * DD + n] = val > 0.0f ? val : 0.0f;
    }
}

// ---------------------------------------------------------------- kernel 3
// out = H + y2[batch_idx]  (float4 grid-stride stream; y2 is L2-resident).
__global__ void k_residual(const float* __restrict__ H, const int* __restrict__ idx,
                           const float* __restrict__ Y2, float* __restrict__ out,
                           int N) {
    const size_t total  = (size_t)N * (DD / 4);   // float4 elements
    const size_t stride = (size_t)gridDim.x * blockDim.x;
    for (size_t i = (size_t)blockIdx.x * blockDim.x + threadIdx.x; i < total; i += stride) {
        const size_t row = i >> 5;                // 32 float4 per row
        const int    c4  = (int)(i & 31);
        const int    g   = idx[row];
        const v4f    h   = __builtin_nontemporal_load((const v4f*)H + i);
        const v4f    yv  = ((const v4f*)Y2)[(size_t)g * 32 + c4];
        const v4f    o   = h + yv;
        __builtin_nontemporal_store(o, (v4f*)out + i);
    }
}

// ---------------------------------------------------------------- launcher
extern "C" void kernel_launch(void* const* d_in, const int* in_sizes, int n_in,
                              void* d_out, int out_size, void* d_ws, size_t ws_size,
                              hipStream_t stream) {
    const float* H    = (const float*)d_in[0];
    const int*   idx  = (const int*)d_in[1];
    const int*   gptr = (const int*)d_in[2];   // num_graphs (device scalar)
    const float* W    = (const float*)d_in[3];
    const float* bias = (const float*)d_in[4];
    float*       out  = (float*)d_out;

    const int N = in_sizes[0] / DD;            // 1,000,000

    float* y  = (float*)d_ws;                  // [GMAX, DD] segment sums
    float* y2 = y + (size_t)GMAX * DD;         // [GMAX, DD] relu(linear)

    // 0) zero accumulator
    k_zero<<<(GMAX * DD + 255) / 256, 256, 0, stream>>>(y, gptr);

    // 1) sorted segment sum: 1024 blocks x 8 waves = 8192 waves, ~123 rows each
    k_segsum<<<1024, 256, 0, stream>>>(H, idx, y, N);

    // 2) WMMA GEMM: (GMAX/16)*(DD/16)=512 tiles, 4 waves/block -> 128 blocks
    k_gemm_relu<<<(GMAX / 16) * (DD / 16) / 4, 128, 0, stream>>>(y, W, bias, y2, gptr);

    // 3) broadcast + residual: streaming, float4
    k_residual<<<4096, 256, 0, stream>>>(H, idx, y2, out, N);
}